// gradientfree_50345606643858
// MI455X (gfx1250) — compile-verified
//
#include <hip/hip_runtime.h>
#include <cstdint>

#define BLOCK 256

// ---------------------------------------------------------------------------
// wave32 / block reductions (deterministic trees, no float atomics)
// ---------------------------------------------------------------------------
__device__ __forceinline__ float wave_reduce_add(float v) {
#pragma unroll
  for (int off = 16; off > 0; off >>= 1)
    v += __shfl_down(v, off, 32);
  return v;
}

__device__ __forceinline__ void block_reduce_store(float v, float* __restrict__ partial) {
  __shared__ float sm[BLOCK / 32];
  v = wave_reduce_add(v);
  const int lane = threadIdx.x & 31;
  const int wid  = threadIdx.x >> 5;
  if (lane == 0) sm[wid] = v;
  __syncthreads();
  if (wid == 0) {
    float w = (lane < (BLOCK / 32)) ? sm[lane] : 0.0f;
    w = wave_reduce_add(w);
    if (lane == 0) partial[blockIdx.x] = w;
  }
}

// ---------------------------------------------------------------------------
// Async staging of one lane's 9-int n_index row into LDS (gfx1250 async path).
// Row lives at global 36*i (4B aligned, legal for b128 in DWORD align mode);
// LDS row padded to 48B so the b128 LDS writes are 16B aligned.
// INST_OFFSET advances BOTH the LDS and global address, so one base each.
// Each thread copies and reads only its own row -> s_wait_asynccnt is enough.
// ---------------------------------------------------------------------------
__device__ __forceinline__ void async_stage_row9(const int* __restrict__ grow,
                                                 int* __restrict__ lds_row) {
  const unsigned ldsoff = (unsigned)(uintptr_t)lds_row;  // low 32 bits = LDS byte offset
  asm volatile(
      "global_load_async_to_lds_b128 %0, %1, off\n\t"
      "global_load_async_to_lds_b128 %0, %1, off offset:16\n\t"
      "global_load_async_to_lds_b32  %0, %1, off offset:32\n\t"
      "s_wait_asynccnt 0x0"
      :
      : "v"(ldsoff), "v"(grow)
      : "memory");
}

// ---------------------------------------------------------------------------
// M==9 fast-path kernels (defined first so the disasm snippet shows them).
// Packed layout:
//   p1[i]    = (x0, x1, ut, ut1)                 16B, one L2 line per gather
//   p2[2i]   = (x0, x1, g0, g1)                  g  = grad1 of ut
//   p2[2i+1] = (gb0, gb1, 0, 0)                  gb = grad1 of ut1
// ---------------------------------------------------------------------------
__global__ void grad1_packed9(const float4* __restrict__ p1,
                              const int* __restrict__ nidx,
                              const float* __restrict__ invm,
                              float4* __restrict__ p2, int N) {
  __shared__ int sidx[BLOCK * 12];
  const int i = blockIdx.x * BLOCK + threadIdx.x;
  if (i >= N) return;
  const float4 self = p1[i];
  int* lds_row = &sidx[threadIdx.x * 12];
  async_stage_row9(nidx + (size_t)i * 9, lds_row);
  float s0 = 0.f, s1 = 0.f, t0 = 0.f, t1 = 0.f;
#pragma unroll
  for (int m = 0; m < 9; ++m) {
    const float4 nb = p1[lds_row[m]];
    const float dx0 = nb.x - self.x, dx1 = nb.y - self.y;
    const float du  = nb.z - self.z, dv  = nb.w - self.w;
    s0 = fmaf(du, dx0, s0); s1 = fmaf(du, dx1, s1);
    t0 = fmaf(dv, dx0, t0); t1 = fmaf(dv, dx1, t1);
  }
  const float4 mm = ((const float4*)invm)[i];  // (m00, m01, m10, m11)
  // g_i = sum_j s_j * inv[j][i]
  p2[2 * i]     = make_float4(self.x, self.y,
                              fmaf(s0, mm.x, s1 * mm.z),
                              fmaf(s0, mm.y, s1 * mm.w));
  p2[2 * i + 1] = make_float4(fmaf(t0, mm.x, t1 * mm.z),
                              fmaf(t0, mm.y, t1 * mm.w), 0.f, 0.f);
}

__global__ void grad2_packed9(const float4* __restrict__ p1,
                              const float4* __restrict__ p2,
                              const float* __restrict__ invm,
                              const float* __restrict__ up,
                              const float* __restrict__ usol,
                              const int* __restrict__ nidx,
                              float* __restrict__ partial, int N) {
  __shared__ int sidx[BLOCK * 12];
  const int i = blockIdx.x * BLOCK + threadIdx.x;
  float contrib = 0.f;
  if (i < N) {
    const float4 a = p2[2 * i];       // x0,x1,g0,g1
    const float4 b = p2[2 * i + 1];   // gb0,gb1,-,-
    int* lds_row = &sidx[threadIdx.x * 12];
    async_stage_row9(nidx + (size_t)i * 9, lds_row);
    float S00 = 0.f, S01 = 0.f, S10 = 0.f, S11 = 0.f;
    float T00 = 0.f, T01 = 0.f, T10 = 0.f, T11 = 0.f;
#pragma unroll
    for (int m = 0; m < 9; ++m) {
      const int k = lds_row[m];
      const float4 na  = p2[2 * k];
      const float4 nbv = p2[2 * k + 1];
      const float dx0 = na.x - a.x, dx1 = na.y - a.y;
      const float dg0 = na.z - a.z, dg1 = na.w - a.w;
      const float db0 = nbv.x - b.x, db1 = nbv.y - b.y;
      S00 = fmaf(dg0, dx0, S00); S01 = fmaf(dg0, dx1, S01);
      S10 = fmaf(dg1, dx0, S10); S11 = fmaf(dg1, dx1, S11);
      T00 = fmaf(db0, dx0, T00); T01 = fmaf(db0, dx1, T01);
      T10 = fmaf(db1, dx0, T10); T11 = fmaf(db1, dx1, T11);
    }
    const float4 mm = ((const float4*)invm)[i];
    // out[j][l] = sum_p S[j][p] * inv[p][l]; keep (0,0) and (1,1)
    const float uxx  = fmaf(S00, mm.x, S01 * mm.z);
    const float uyy  = fmaf(S10, mm.y, S11 * mm.w);
    const float uxx1 = fmaf(T00, mm.x, T01 * mm.z);
    const float uyy1 = fmaf(T10, mm.y, T11 * mm.w);
    const float4 s = p1[i];
    const float uti = s.z, u1i = s.w;
    const float inner = 0.01f * (uxx + uyy) + uti - uti * uti * uti
                      + 0.01f * (uxx1 + uyy1) + u1i - u1i * u1i * u1i;
    const float f = (u1i - uti) - 0.01f * inner;   // 0.5*0.02 = 0.01
    const float e = up[i] - usol[i];
    contrib = fmaf(e, e, 4.f * f * f);             // loss_u term folded in
  }
  block_reduce_store(contrib, partial);
}

// ---------------------------------------------------------------------------
// p1 packing kernel
// ---------------------------------------------------------------------------
__global__ void pack_kernel(const float* __restrict__ x,
                            const float* __restrict__ ut,
                            const float* __restrict__ ut1,
                            float4* __restrict__ p1, int N) {
  const int i = blockIdx.x * BLOCK + threadIdx.x;
  if (i < N) {
    const float2 xi = ((const float2*)x)[i];
    p1[i] = make_float4(xi.x, xi.y, ut[i], ut1[i]);
  }
}

// ---------------------------------------------------------------------------
// Generic-M packed kernels (runtime neighbor count)
// ---------------------------------------------------------------------------
__global__ void grad1_packed_gen(const float4* __restrict__ p1,
                                 const int* __restrict__ nidx,
                                 const float* __restrict__ invm,
                                 float4* __restrict__ p2, int N, int Mrt) {
  const int i = blockIdx.x * BLOCK + threadIdx.x;
  if (i >= N) return;
  const float4 self = p1[i];
  const int* __restrict__ row = nidx + (size_t)i * Mrt;
  float s0 = 0.f, s1 = 0.f, t0 = 0.f, t1 = 0.f;
  for (int m = 0; m < Mrt; ++m) {
    const float4 nb = p1[row[m]];
    const float dx0 = nb.x - self.x, dx1 = nb.y - self.y;
    const float du  = nb.z - self.z, dv  = nb.w - self.w;
    s0 = fmaf(du, dx0, s0); s1 = fmaf(du, dx1, s1);
    t0 = fmaf(dv, dx0, t0); t1 = fmaf(dv, dx1, t1);
  }
  const float4 mm = ((const float4*)invm)[i];
  p2[2 * i]     = make_float4(self.x, self.y,
                              fmaf(s0, mm.x, s1 * mm.z),
                              fmaf(s0, mm.y, s1 * mm.w));
  p2[2 * i + 1] = make_float4(fmaf(t0, mm.x, t1 * mm.z),
                              fmaf(t0, mm.y, t1 * mm.w), 0.f, 0.f);
}

__global__ void grad2_packed_gen(const float4* __restrict__ p1,
                                 const float4* __restrict__ p2,
                                 const float* __restrict__ invm,
                                 const float* __restrict__ up,
                                 const float* __restrict__ usol,
                                 const int* __restrict__ nidx,
                                 float* __restrict__ partial, int N, int Mrt) {
  const int i = blockIdx.x * BLOCK + threadIdx.x;
  float contrib = 0.f;
  if (i < N) {
    const float4 a = p2[2 * i];
    const float4 b = p2[2 * i + 1];
    const int* __restrict__ row = nidx + (size_t)i * Mrt;
    float S00 = 0.f, S01 = 0.f, S10 = 0.f, S11 = 0.f;
    float T00 = 0.f, T01 = 0.f, T10 = 0.f, T11 = 0.f;
    for (int m = 0; m < Mrt; ++m) {
      const int k = row[m];
      const float4 na  = p2[2 * k];
      const float4 nbv = p2[2 * k + 1];
      const float dx0 = na.x - a.x, dx1 = na.y - a.y;
      const float dg0 = na.z - a.z, dg1 = na.w - a.w;
      const float db0 = nbv.x - b.x, db1 = nbv.y - b.y;
      S00 = fmaf(dg0, dx0, S00); S01 = fmaf(dg0, dx1, S01);
      S10 = fmaf(dg1, dx0, S10); S11 = fmaf(dg1, dx1, S11);
      T00 = fmaf(db0, dx0, T00); T01 = fmaf(db0, dx1, T01);
      T10 = fmaf(db1, dx0, T10); T11 = fmaf(db1, dx1, T11);
    }
    const float4 mm = ((const float4*)invm)[i];
    const float uxx  = fmaf(S00, mm.x, S01 * mm.z);
    const float uyy  = fmaf(S10, mm.y, S11 * mm.w);
    const float uxx1 = fmaf(T00, mm.x, T01 * mm.z);
    const float uyy1 = fmaf(T10, mm.y, T11 * mm.w);
    const float4 s = p1[i];
    const float uti = s.z, u1i = s.w;
    const float inner = 0.01f * (uxx + uyy) + uti - uti * uti * uti
                      + 0.01f * (uxx1 + uyy1) + u1i - u1i * u1i * u1i;
    const float f = (u1i - uti) - 0.01f * inner;
    const float e = up[i] - usol[i];
    contrib = fmaf(e, e, 4.f * f * f);
  }
  block_reduce_store(contrib, partial);
}

// ---------------------------------------------------------------------------
// Basic (low-workspace) fallback: same math, unpacked gathers.
// ---------------------------------------------------------------------------
template <int MT>
__global__ void grad1_basic(const float* __restrict__ x,
                            const float* __restrict__ ut,
                            const float* __restrict__ ut1,
                            const int* __restrict__ nidx,
                            const float* __restrict__ invm,
                            float2* __restrict__ g, float2* __restrict__ gb,
                            int N, int Mrt) {
  const int i = blockIdx.x * BLOCK + threadIdx.x;
  if (i >= N) return;
  const int M = MT ? MT : Mrt;
  const float2 xi = ((const float2*)x)[i];
  const float uti = ut[i], u1i = ut1[i];
  const int* __restrict__ row = nidx + (size_t)i * M;
  float s0 = 0.f, s1 = 0.f, t0 = 0.f, t1 = 0.f;
  auto body = [&](int m) {
    const int k = row[m];
    const float2 xn = ((const float2*)x)[k];
    const float dx0 = xn.x - xi.x, dx1 = xn.y - xi.y;
    const float du = ut[k] - uti, dv = ut1[k] - u1i;
    s0 = fmaf(du, dx0, s0); s1 = fmaf(du, dx1, s1);
    t0 = fmaf(dv, dx0, t0); t1 = fmaf(dv, dx1, t1);
  };
  if constexpr (MT > 0) {
#pragma unroll
    for (int m = 0; m < MT; ++m) body(m);
  } else {
    for (int m = 0; m < Mrt; ++m) body(m);
  }
  const float4 mm = ((const float4*)invm)[i];
  g[i]  = make_float2(fmaf(s0, mm.x, s1 * mm.z), fmaf(s0, mm.y, s1 * mm.w));
  gb[i] = make_float2(fmaf(t0, mm.x, t1 * mm.z), fmaf(t0, mm.y, t1 * mm.w));
}

template <int MT>
__global__ void grad2_basic(const float* __restrict__ x,
                            const float2* __restrict__ g,
                            const float2* __restrict__ gb,
                            const float* __restrict__ ut,
                            const float* __restrict__ ut1,
                            const float* __restrict__ up,
                            const float* __restrict__ usol,
                            const int* __restrict__ nidx,
                            const float* __restrict__ invm,
                            float* __restrict__ partial, int N, int Mrt) {
  const int i = blockIdx.x * BLOCK + threadIdx.x;
  float contrib = 0.f;
  if (i < N) {
    const int M = MT ? MT : Mrt;
    const float2 xi = ((const float2*)x)[i];
    const float2 gi = g[i];
    const float2 bi = gb[i];
    const int* __restrict__ row = nidx + (size_t)i * M;
    float S00 = 0.f, S01 = 0.f, S10 = 0.f, S11 = 0.f;
    float T00 = 0.f, T01 = 0.f, T10 = 0.f, T11 = 0.f;
    auto body = [&](int m) {
      const int k = row[m];
      const float2 xn = ((const float2*)x)[k];
      const float2 gn = g[k];
      const float2 bn = gb[k];
      const float dx0 = xn.x - xi.x, dx1 = xn.y - xi.y;
      const float dg0 = gn.x - gi.x, dg1 = gn.y - gi.y;
      const float db0 = bn.x - bi.x, db1 = bn.y - bi.y;
      S00 = fmaf(dg0, dx0, S00); S01 = fmaf(dg0, dx1, S01);
      S10 = fmaf(dg1, dx0, S10); S11 = fmaf(dg1, dx1, S11);
      T00 = fmaf(db0, dx0, T00); T01 = fmaf(db0, dx1, T01);
      T10 = fmaf(db1, dx0, T10); T11 = fmaf(db1, dx1, T11);
    };
    if constexpr (MT > 0) {
#pragma unroll
      for (int m = 0; m < MT; ++m) body(m);
    } else {
      for (int m = 0; m < Mrt; ++m) body(m);
    }
    const float4 mm = ((const float4*)invm)[i];
    const float uxx  = fmaf(S00, mm.x, S01 * mm.z);
    const float uyy  = fmaf(S10, mm.y, S11 * mm.w);
    const float uxx1 = fmaf(T00, mm.x, T01 * mm.z);
    const float uyy1 = fmaf(T10, mm.y, T11 * mm.w);
    const float uti = ut[i], u1i = ut1[i];
    const float inner = 0.01f * (uxx + uyy) + uti - uti * uti * uti
                      + 0.01f * (uxx1 + uyy1) + u1i - u1i * u1i * u1i;
    const float f = (u1i - uti) - 0.01f * inner;
    const float e = up[i] - usol[i];
    contrib = fmaf(e, e, 4.f * f * f);
  }
  block_reduce_store(contrib, partial);
}

// ---------------------------------------------------------------------------
// Final fixed-order reduction of per-block partials -> scalar loss
// ---------------------------------------------------------------------------
__global__ void final_reduce(const float* __restrict__ partial, int nblk,
                             float* __restrict__ out) {
  float s = 0.f;
  for (int j = threadIdx.x; j < nblk; j += BLOCK) s += partial[j];
  __shared__ float sm[BLOCK / 32];
  s = wave_reduce_add(s);
  const int lane = threadIdx.x & 31;
  const int wid  = threadIdx.x >> 5;
  if (lane == 0) sm[wid] = s;
  __syncthreads();
  if (wid == 0) {
    float w = (lane < (BLOCK / 32)) ? sm[lane] : 0.0f;
    w = wave_reduce_add(w);
    if (lane == 0) out[0] = w;
  }
}

// ---------------------------------------------------------------------------
extern "C" void kernel_launch(void* const* d_in, const int* in_sizes, int n_in,
                              void* d_out, int out_size, void* d_ws, size_t ws_size,
                              hipStream_t stream) {
  const float* up   = (const float*)d_in[0];
  const float* usol = (const float*)d_in[1];
  const float* ut   = (const float*)d_in[2];
  const float* x    = (const float*)d_in[3];
  const float* ut1  = (const float*)d_in[4];
  const int*   nidx = (const int*)d_in[5];
  const float* invm = (const float*)d_in[6];
  float* out = (float*)d_out;

  const int N = in_sizes[0];
  const int M = (N > 0) ? (in_sizes[5] / N) : 0;
  const int nblk = (N + BLOCK - 1) / BLOCK;

  char* ws = (char*)d_ws;
  const size_t packedNeed = (size_t)N * 16 + (size_t)N * 32 + (size_t)nblk * 4;

  if (ws_size >= packedNeed) {
    float4* p1      = (float4*)ws;
    float4* p2      = (float4*)(ws + (size_t)N * 16);
    float*  partial = (float*)(ws + (size_t)N * 16 + (size_t)N * 32);
    pack_kernel<<<nblk, BLOCK, 0, stream>>>(x, ut, ut1, p1, N);
    if (M == 9) {
      grad1_packed9<<<nblk, BLOCK, 0, stream>>>(p1, nidx, invm, p2, N);
      grad2_packed9<<<nblk, BLOCK, 0, stream>>>(p1, p2, invm, up, usol, nidx,
                                                partial, N);
    } else {
      grad1_packed_gen<<<nblk, BLOCK, 0, stream>>>(p1, nidx, invm, p2, N, M);
      grad2_packed_gen<<<nblk, BLOCK, 0, stream>>>(p1, p2, invm, up, usol, nidx,
                                                   partial, N, M);
    }
    final_reduce<<<1, BLOCK, 0, stream>>>(partial, nblk, out);
  } else {
    float2* g       = (float2*)ws;
    float2* gb      = (float2*)(ws + (size_t)N * 8);
    float*  partial = (float*)(ws + (size_t)N * 16);
    if (M == 9) {
      grad1_basic<9><<<nblk, BLOCK, 0, stream>>>(x, ut, ut1, nidx, invm, g, gb, N, M);
      grad2_basic<9><<<nblk, BLOCK, 0, stream>>>(x, g, gb, ut, ut1, up, usol, nidx,
                                                 invm, partial, N, M);
    } else {
      grad1_basic<0><<<nblk, BLOCK, 0, stream>>>(x, ut, ut1, nidx, invm, g, gb, N, M);
      grad2_basic<0><<<nblk, BLOCK, 0, stream>>>(x, g, gb, ut, ut1, up, usol, nidx,
                                                 invm, partial, N, M);
    }
    final_reduce<<<1, BLOCK, 0, stream>>>(partial, nblk, out);
  }
}